// Unpool_41480794144815
// MI455X (gfx1250) — compile-verified
//
#include <hip/hip_runtime.h>

typedef float v4f __attribute__((ext_vector_type(4)));
typedef int   v4i __attribute__((ext_vector_type(4)));

#ifndef __has_builtin
#define __has_builtin(x) 0
#endif

#define Bdim 32
#define Cdim 64
#define Ndim 128
#define Mdim 128

#define THREADS 256
#define ITERS 16

typedef __attribute__((address_space(1))) v4i  as1_v4i;   // global int4*
typedef __attribute__((address_space(3))) v4i  as3_v4i;   // LDS int4*
typedef __attribute__((address_space(3))) char as3_char;

// ---- CDNA5 async global->LDS copy (ASYNCcnt-tracked) -----------------------
__device__ __forceinline__ void async_copy_b128(const float* gsrc, v4f* lds_dst) {
#if __has_builtin(__builtin_amdgcn_global_load_async_to_lds_b128)
  __builtin_amdgcn_global_load_async_to_lds_b128(
      (as1_v4i*)gsrc, (as3_v4i*)lds_dst, /*offset=*/0, /*cpol=*/0);
#else
  // VDST = LDS byte address, VADDR = 64-bit global address, saddr off
  unsigned loff = (unsigned)(unsigned long long)(as3_char*)lds_dst;
  asm volatile("global_load_async_to_lds_b128 %0, %1, off"
               :: "v"(loff), "v"(gsrc)
               : "memory");
#endif
}

template <int N>
__device__ __forceinline__ void wait_async() {
#if __has_builtin(__builtin_amdgcn_s_wait_asynccnt)
  __builtin_amdgcn_s_wait_asynccnt(N);
#else
  asm volatile("s_wait_asynccnt %0" :: "n"(N) : "memory");
#endif
}

// ---- Kernel B: streaming 2x2 scatter, async-LDS pipelined, NT b128 stores --
// (defined first so its device asm appears at the top of the disassembly)
__global__ __launch_bounds__(THREADS) void unpool_scatter(const float* __restrict__ s,
                                                          const unsigned char* __restrict__ zetaT,
                                                          float* __restrict__ out) {
  __shared__ v4f buf[2][THREADS];
  const unsigned tid = threadIdx.x;
  const unsigned g0  = blockIdx.x * (THREADS * ITERS);   // float4-group base

  async_copy_b128(s + (size_t)(g0 + tid) * 4, &buf[0][tid]);

#pragma unroll 1
  for (int it = 0; it < ITERS; ++it) {
    const int cur = it & 1;
    if (it + 1 < ITERS) {
      async_copy_b128(s + (size_t)(g0 + (unsigned)(it + 1) * THREADS + tid) * 4,
                      &buf[cur ^ 1][tid]);
      wait_async<1>();   // current buffer complete, prefetch in flight
    } else {
      wait_async<0>();
    }
    v4f s4 = buf[cur][tid];                              // ds_read_b128, own slot only

    const unsigned g   = g0 + (unsigned)it * THREADS + tid;
    const unsigned j0  = (g & 31u) << 2;                 // (g*4) % 128
    const unsigned row = g >> 5;                         // (b*C + c)*n + i
    const unsigned i   = row & (Ndim - 1);
    const unsigned bc  = row >> 7;
    const unsigned c   = bc & (Cdim - 1);

    const unsigned zt = (c * Ndim + i) * Mdim + j0;      // 4 contiguous zeta bytes
    const unsigned zb = *(const unsigned*)(zetaT + zt);  // cacheable: reused 32x over b

    const unsigned p0 = zb & 3u, p1 = (zb >> 8) & 3u,
                   p2 = (zb >> 16) & 3u, p3 = (zb >> 24) & 3u;

    v4f top0 = {0.f, 0.f, 0.f, 0.f}, top1 = top0, bot0 = top0, bot1 = top0;
    // p = px*2 + py ; row 2i+px, col 2j+py  (branch-free selects)
    top0.x = (p0 == 0u) ? s4.x : 0.f;  top0.y = (p0 == 1u) ? s4.x : 0.f;
    bot0.x = (p0 == 2u) ? s4.x : 0.f;  bot0.y = (p0 == 3u) ? s4.x : 0.f;
    top0.z = (p1 == 0u) ? s4.y : 0.f;  top0.w = (p1 == 1u) ? s4.y : 0.f;
    bot0.z = (p1 == 2u) ? s4.y : 0.f;  bot0.w = (p1 == 3u) ? s4.y : 0.f;
    top1.x = (p2 == 0u) ? s4.z : 0.f;  top1.y = (p2 == 1u) ? s4.z : 0.f;
    bot1.x = (p2 == 2u) ? s4.z : 0.f;  bot1.y = (p2 == 3u) ? s4.z : 0.f;
    top1.z = (p3 == 0u) ? s4.w : 0.f;  top1.w = (p3 == 1u) ? s4.w : 0.f;
    bot1.z = (p3 == 2u) ? s4.w : 0.f;  bot1.w = (p3 == 3u) ? s4.w : 0.f;

    // out[bc][2i][2j0..], out[bc][2i+1][2j0..] ; 2n*2m = 65536, 2m = 256
    const unsigned obase = (bc << 16) + ((2u * i) << 8) + 2u * j0;
    __builtin_nontemporal_store(top0, (v4f*)(out + obase));
    __builtin_nontemporal_store(top1, (v4f*)(out + obase) + 1);
    __builtin_nontemporal_store(bot0, (v4f*)(out + obase + 2 * Mdim));
    __builtin_nontemporal_store(bot1, (v4f*)(out + obase + 2 * Mdim) + 1);
  }
}

// ---- Kernel A: categorical sample per (c,i,j), stored transposed [C,n,m] ---
__global__ __launch_bounds__(256) void zeta_sample(const float* __restrict__ etas,
                                                   unsigned char* __restrict__ zetaT) {
  const unsigned t  = blockIdx.x * 256u + threadIdx.x;   // t = (c*n + i)*m + j
  const unsigned j  = t & (Mdim - 1);
  const unsigned ci = t >> 7;
  const unsigned i  = ci & (Ndim - 1);
  const unsigned c  = ci >> 7;
  const unsigned r  = (i * Mdim + j) * Cdim + c;         // reference row order (i,j,k)

  v4f w = __builtin_nontemporal_load((const v4f*)etas + r);  // streaming, one-shot

  // deterministic stateless RNG (splitmix64 of row index, seed 42)
  unsigned long long x = (unsigned long long)r * 0x9E3779B97F4A7C15ull
                       + 42ull * 0xDA942042E4DD58B5ull;
  x = (x ^ (x >> 30)) * 0xBF58476D1CE4E5B9ull;
  x = (x ^ (x >> 27)) * 0x94D049BB133111EBull;
  x ^= x >> 31;
  float u = (float)(unsigned)((x >> 40) & 0xFFFFFFull) * (1.0f / 16777216.0f);

  float w0 = fmaxf(w.x, 0.f), w1 = fmaxf(w.y, 0.f);
  float w2 = fmaxf(w.z, 0.f), w3 = fmaxf(w.w, 0.f);
  float a1 = w0, a2 = w0 + w1, a3 = a2 + w2;
  float target = u * (a3 + w3);
  unsigned p = (unsigned)(target >= a1) + (unsigned)(target >= a2) +
               (unsigned)(target >= a3);
  zetaT[t] = (unsigned char)p;
}

extern "C" void kernel_launch(void* const* d_in, const int* in_sizes, int n_in,
                              void* d_out, int out_size, void* d_ws, size_t ws_size,
                              hipStream_t stream) {
  (void)in_sizes; (void)n_in; (void)out_size; (void)ws_size;
  const float* s    = (const float*)d_in[0];
  const float* etas = (const float*)d_in[1];
  float* out        = (float*)d_out;
  unsigned char* zetaT = (unsigned char*)d_ws;   // C*n*m = 1 MB

  zeta_sample<<<(Cdim * Ndim * Mdim) / 256, 256, 0, stream>>>(etas, zetaT);

  const unsigned groups = (Bdim * Cdim * Ndim * Mdim) / 4;   // 8,388,608 float4s
  unpool_scatter<<<groups / (THREADS * ITERS), THREADS, 0, stream>>>(s, zetaT, out);
}